// EdgeMessage_48696339202376
// MI455X (gfx1250) — compile-verified
//
#include <hip/hip_runtime.h>

// EdgeMessage fused as one GEMM:  out[E,32] = P[E,544] x B[544,32]
//   P[e, k*32+h] = ev_aug[e,k] * h_w[e,h]   (ev_aug[:,16] = 1.0 -> bias term)
//   B[k*32+h, m] = (k<16) ? W[m*32+h, k] : b[m*32+h]
// A fragments are synthesized in registers (no HBM traffic for P); B (68KB)
// is staged once per workgroup into dynamic LDS (CDNA5: 320KB/WGP).
// Native f32 WMMA (V_WMMA_F32_16X16X4_F32) for bit-exact f32 math.
// v2: each wave carries TWO 16-edge M-tiles so every B-fragment
//     ds_load_2addr_b64 feeds 4 independent WMMAs (better XDL utilization,
//     half the LDS traffic and dscnt waits per WMMA).

typedef __attribute__((ext_vector_type(2))) float v2f;
typedef __attribute__((ext_vector_type(4))) float v4f;
typedef __attribute__((ext_vector_type(8))) float v8f;

namespace {
constexpr int kEdgeDim = 16;                          // e_vw feature dim
constexpr int kHidden  = 32;                          // h_w feature dim
constexpr int kMsg     = 32;                          // output channels
constexpr int kKTot    = (kEdgeDim + 1) * kHidden;    // 544 (bias folded in)
constexpr int kNPair   = kKTot / 2;                   // 272 K-pairs
constexpr int kBLFloats = kNPair * kMsg * 2;          // 17408 floats = 69632 B
constexpr int kWavesPerWG = 8;
constexpr int kThreads = kWavesPerWG * 32;            // wave32
constexpr int kHWFloatsPerWave = 32 * kHidden;        // 32-edge (2 M-tiles)
constexpr int kSmemBytes = (kBLFloats + kWavesPerWG * kHWFloatsPerWave) * 4; // 102400
}

extern "C" __global__ __launch_bounds__(kThreads)
void edge_message_wmma_f32(const float* __restrict__ h_w,
                           const float* __restrict__ e_vw,
                           const float* __restrict__ W,
                           const float* __restrict__ bias,
                           float* __restrict__ out,
                           int E)
{
    extern __shared__ float smem[];
    float* BL = smem;                       // [272 pairs][32 n][2] pair-interleaved
    float* HW = smem + kBLFloats;           // [8 waves][32 edges][32 h] row-major

    const int tid = threadIdx.x;

    // ---- stage B = fused(W, bias) into LDS, pair-interleaved so that one
    //      ds_load_b64 yields the {K=j, K=j+1} B-fragment pair per lane ----
    for (int i = tid; i < kNPair * kMsg; i += kThreads) {
        const int p = i >> 5;               // K-pair index, j = 2p
        const int n = i & 31;               // output channel
        const int j = p << 1;
        const int k = j >> 5;               // 0..16
        const int h = j & 31;               // even
        float x0, x1;
        if (k < kEdgeDim) {
            x0 = W[(n * kHidden + h)     * kEdgeDim + k];
            x1 = W[(n * kHidden + h + 1) * kEdgeDim + k];
        } else {                            // bias slot (ev_aug = 1.0)
            x0 = bias[n * kHidden + h];
            x1 = bias[n * kHidden + h + 1];
        }
        BL[i * 2]     = x0;
        BL[i * 2 + 1] = x1;
    }
    __syncthreads();   // only sync needed: per-tile staging below is wave-private

    const int wave = tid >> 5;
    const int lane = tid & 31;
    const int half = lane >> 4;             // A/B lane-half -> K = 2*half + vgpr
    const int er   = lane & 15;             // A: M-row / B: N-col within tile
    float* HWw = HW + wave * kHWFloatsPerWave;

    const int npairs  = (E + 31) >> 5;      // pairs of 16-edge M-tiles
    const int gw0     = blockIdx.x * kWavesPerWG + wave;
    const int gstride = gridDim.x * kWavesPerWG;

    for (int tp = gw0; tp < npairs; tp += gstride) {
        const int e0 = tp << 5;             // 32 edges per wave iteration

        // stage this wave's 32x32 h_w block into LDS (each lane: one row);
        // LDS ops from one wave are in-order, so no extra sync is needed.
        {
            int se = e0 + lane; if (se > E - 1) se = E - 1;
            const v4f* src = (const v4f*)(h_w + (size_t)se * kHidden);
            v4f* dst = (v4f*)(HWw + lane * kHidden);
            #pragma unroll
            for (int q = 0; q < 8; ++q) dst[q] = src[q];
        }

        // per-lane e_vw rows for both M-tiles (lane halves duplicate row er)
        int ea = e0 + er;      if (ea > E - 1) ea = E - 1;
        int eb = e0 + 16 + er; if (eb > E - 1) eb = E - 1;
        float ev0[kEdgeDim + 1], ev1[kEdgeDim + 1];
        {
            const v4f* sa = (const v4f*)(e_vw + (size_t)ea * kEdgeDim);
            v4f t0 = sa[0], t1 = sa[1], t2 = sa[2], t3 = sa[3];
            ev0[0]=t0.x;  ev0[1]=t0.y;  ev0[2]=t0.z;  ev0[3]=t0.w;
            ev0[4]=t1.x;  ev0[5]=t1.y;  ev0[6]=t1.z;  ev0[7]=t1.w;
            ev0[8]=t2.x;  ev0[9]=t2.y;  ev0[10]=t2.z; ev0[11]=t2.w;
            ev0[12]=t3.x; ev0[13]=t3.y; ev0[14]=t3.z; ev0[15]=t3.w;
            ev0[16]=1.0f;
            const v4f* sb = (const v4f*)(e_vw + (size_t)eb * kEdgeDim);
            v4f u0 = sb[0], u1 = sb[1], u2 = sb[2], u3 = sb[3];
            ev1[0]=u0.x;  ev1[1]=u0.y;  ev1[2]=u0.z;  ev1[3]=u0.w;
            ev1[4]=u1.x;  ev1[5]=u1.y;  ev1[6]=u1.z;  ev1[7]=u1.w;
            ev1[8]=u2.x;  ev1[9]=u2.y;  ev1[10]=u2.z; ev1[11]=u2.w;
            ev1[12]=u3.x; ev1[13]=u3.y; ev1[14]=u3.z; ev1[15]=u3.w;
            ev1[16]=1.0f;
        }

        // hoist the h_w pairs each lane needs (ISA 16x4 f32 A layout:
        // lane holds A[M=er, K=2*half+{0,1}] -> h = hc*4 + 2*half + {0,1})
        v2f hwp0[8], hwp1[8];
        #pragma unroll
        for (int hc = 0; hc < 8; ++hc) {
            hwp0[hc] = *(const v2f*)(HWw + er        * kHidden + hc * 4 + 2 * half);
            hwp1[hc] = *(const v2f*)(HWw + (16 + er) * kHidden + hc * 4 + 2 * half);
        }

        v8f acc00 = {0,0,0,0,0,0,0,0};      // tile0, N = 0..15
        v8f acc01 = {0,0,0,0,0,0,0,0};      // tile0, N = 16..31
        v8f acc10 = {0,0,0,0,0,0,0,0};      // tile1, N = 0..15
        v8f acc11 = {0,0,0,0,0,0,0,0};      // tile1, N = 16..31

        #pragma unroll
        for (int k0 = 0; k0 < kEdgeDim + 1; ++k0) {     // 17 K-groups of 32
            const float evk0 = ev0[k0];                 // compile-time indices
            const float evk1 = ev1[k0];
            #pragma unroll
            for (int hc = 0; hc < 8; ++hc) {            // K-chunk of 4 per WMMA
                v2f a0, a1;
                a0.x = hwp0[hc].x * evk0;               // A = ev ⊗ h_w on the fly
                a0.y = hwp0[hc].y * evk0;
                a1.x = hwp1[hc].x * evk1;
                a1.y = hwp1[hc].y * evk1;
                const int p0 = k0 * 16 + hc * 2 + half; // K-pair for this lane
                const v2f* bp = (const v2f*)BL + p0 * kMsg;
                v2f b0 = bp[er];                        // one ds_load_2addr_b64
                v2f b1 = bp[16 + er];                   // feeds 4 WMMAs
                acc00 = __builtin_amdgcn_wmma_f32_16x16x4_f32(
                    false, a0, false, b0, (short)0, acc00, false, false);
                acc01 = __builtin_amdgcn_wmma_f32_16x16x4_f32(
                    false, a0, false, b1, (short)0, acc01, false, false);
                acc10 = __builtin_amdgcn_wmma_f32_16x16x4_f32(
                    false, a1, false, b0, (short)0, acc10, false, false);
                acc11 = __builtin_amdgcn_wmma_f32_16x16x4_f32(
                    false, a1, false, b1, (short)0, acc11, false, false);
            }
        }

        // D layout: N = lane%16, M = r + 8*half
        #pragma unroll
        for (int r = 0; r < 8; ++r) {
            const int row0 = e0 + r + 8 * half;         // tile0 edge row
            if (row0 < E) {
                out[(size_t)row0 * kMsg + er]      = acc00[r];
                out[(size_t)row0 * kMsg + 16 + er] = acc01[r];
            }
            const int row1 = e0 + 16 + r + 8 * half;    // tile1 edge row
            if (row1 < E) {
                out[(size_t)row1 * kMsg + er]      = acc10[r];
                out[(size_t)row1 * kMsg + 16 + er] = acc11[r];
            }
        }
    }
}

extern "C" void kernel_launch(void* const* d_in, const int* in_sizes, int n_in,
                              void* d_out, int out_size, void* d_ws, size_t ws_size,
                              hipStream_t stream) {
    const float* h_w  = (const float*)d_in[0];
    const float* e_vw = (const float*)d_in[1];
    const float* W    = (const float*)d_in[2];
    const float* bias = (const float*)d_in[3];
    float* out = (float*)d_out;

    const int E = in_sizes[0] / kHidden;
    const int npairs = (E + 31) / 32;
    int nwg = (npairs + kWavesPerWG - 1) / kWavesPerWG;
    if (nwg > 768) nwg = 768;     // grid-stride: amortize 68KB B-staging per WG
    if (nwg < 1) nwg = 1;

    edge_message_wmma_f32<<<nwg, kThreads, kSmemBytes, stream>>>(
        h_w, e_vw, W, bias, out, E);
}